// CLoss_inout_53712861004462
// MI455X (gfx1250) — compile-verified
//
#include <hip/hip_runtime.h>

typedef __attribute__((ext_vector_type(2))) float v2f;
typedef __attribute__((ext_vector_type(8))) float v8f;

#define THREADS 256
#define WAVES_PER_BLOCK (THREADS / 32)
#define UNROLL 4

// Streaming row-dot partial-sum kernel. Bulk work: 8 coalesced b128 loads in
// flight per iteration + fp32 FMA into 4 independent accumulators (pure
// bandwidth bound, uniform scalar trip count -> no exec-mask churn).
// Wave reduction via V_WMMA_F32_16X16X4_F32 (exact fp32 adds) vs ones-B.
__global__ void __launch_bounds__(THREADS)
dot_partial_kernel(const float4* __restrict__ a,
                   const float4* __restrict__ b,
                   float* __restrict__ block_partials,
                   int nvec, int iters) {
  const int tid = blockIdx.x * blockDim.x + threadIdx.x;
  const int stride = gridDim.x * blockDim.x;

  float acc0 = 0.0f, acc1 = 0.0f, acc2 = 0.0f, acc3 = 0.0f;
  int i = tid;

  // Main loop: uniform trip count (scalar), 4x unrolled, 8 b128 loads/iter.
  for (int it = 0; it < iters; ++it) {
    float4 x0 = a[i];
    float4 y0 = b[i];
    float4 x1 = a[i + stride];
    float4 y1 = b[i + stride];
    float4 x2 = a[i + 2 * stride];
    float4 y2 = b[i + 2 * stride];
    float4 x3 = a[i + 3 * stride];
    float4 y3 = b[i + 3 * stride];

    acc0 = fmaf(x0.x, y0.x, acc0);
    acc0 = fmaf(x0.y, y0.y, acc0);
    acc0 = fmaf(x0.z, y0.z, acc0);
    acc0 = fmaf(x0.w, y0.w, acc0);

    acc1 = fmaf(x1.x, y1.x, acc1);
    acc1 = fmaf(x1.y, y1.y, acc1);
    acc1 = fmaf(x1.z, y1.z, acc1);
    acc1 = fmaf(x1.w, y1.w, acc1);

    acc2 = fmaf(x2.x, y2.x, acc2);
    acc2 = fmaf(x2.y, y2.y, acc2);
    acc2 = fmaf(x2.z, y2.z, acc2);
    acc2 = fmaf(x2.w, y2.w, acc2);

    acc3 = fmaf(x3.x, y3.x, acc3);
    acc3 = fmaf(x3.y, y3.y, acc3);
    acc3 = fmaf(x3.z, y3.z, acc3);
    acc3 = fmaf(x3.w, y3.w, acc3);

    i += UNROLL * stride;
  }

  // Tail (runs 0..4 strided iterations; empty when sizes divide evenly).
  for (; i < nvec; i += stride) {
    float4 x = a[i];
    float4 y = b[i];
    acc0 = fmaf(x.x, y.x, acc0);
    acc0 = fmaf(x.y, y.y, acc0);
    acc0 = fmaf(x.z, y.z, acc0);
    acc0 = fmaf(x.w, y.w, acc0);
  }

  float acc = (acc0 + acc1) + (acc2 + acc3);

  // --- Wave32 reduction via WMMA f32 16x16x4 ---
  // A (16x4, 2 VGPRs): lane l<16 holds (K0,K1)=(acc,0); lane l>=16 holds
  // (K2,K3)=(acc,0). B (4x16) = ones. D[m,n] = acc(m) + acc(m+16) for all n.
  v2f A;
  A.x = acc;
  A.y = 0.0f;
  v2f B;
  B.x = 1.0f;
  B.y = 1.0f;
  v8f C = {};
  C = __builtin_amdgcn_wmma_f32_16x16x4_f32(
      /*neg_a=*/false, A, /*neg_b=*/false, B,
      /*c_mod=*/(short)0, C, /*reuse_a=*/false, /*reuse_b=*/false);
  // Half0 lanes hold rowsums M=0..7 across the 8 C VGPRs; half1 holds
  // M=8..15. Sum the 8 VGPRs, then add across halves with one xor-shuffle.
  float s = C[0] + C[1] + C[2] + C[3] + C[4] + C[5] + C[6] + C[7];
  s += __shfl_xor(s, 16, 32);

  __shared__ float lds[WAVES_PER_BLOCK];
  const int lane = threadIdx.x & 31;
  const int wave = threadIdx.x >> 5;
  if (lane == 0) lds[wave] = s;
  __syncthreads();
  if (threadIdx.x == 0) {
    float t = 0.0f;
#pragma unroll
    for (int w = 0; w < WAVES_PER_BLOCK; ++w) t += lds[w];
    block_partials[blockIdx.x] = t;
  }
}

// Deterministic single-block reduction of block partials + epilogue.
__global__ void __launch_bounds__(THREADS)
finalize_kernel(const float* __restrict__ partials, int nparts,
                const int* __restrict__ flip, float* __restrict__ out,
                float invN) {
  __shared__ float lds[THREADS];
  float s = 0.0f;
  for (int i = threadIdx.x; i < nparts; i += THREADS) s += partials[i];
  lds[threadIdx.x] = s;
  __syncthreads();
  for (int off = THREADS / 2; off > 0; off >>= 1) {
    if (threadIdx.x < off) lds[threadIdx.x] += lds[threadIdx.x + off];
    __syncthreads();
  }
  if (threadIdx.x == 0) {
    float mean_sim = lds[0] * invN;
    out[0] = flip[0] ? (mean_sim + 1.0f) : (1.0f - mean_sim);
  }
}

extern "C" void kernel_launch(void* const* d_in, const int* in_sizes, int n_in,
                              void* d_out, int out_size, void* d_ws, size_t ws_size,
                              hipStream_t stream) {
  const float* A = (const float*)d_in[0];
  const float* B = (const float*)d_in[1];
  const int* flip = (const int*)d_in[2];
  float* out = (float*)d_out;
  float* partials = (float*)d_ws;

  const long total_elems = (long)in_sizes[0];   // N * D
  const int D = 128;
  const long n_rows = total_elems / D;
  const int nvec = (int)(total_elems / 4);      // float4 count (fits in int32)

  int nblocks = 4096;
  const size_t ws_cap = ws_size / sizeof(float);
  if ((size_t)nblocks > ws_cap) nblocks = (int)ws_cap;
  if (nblocks < 1) nblocks = 1;

  const int total_threads = nblocks * THREADS;
  const int iters = nvec / (total_threads * UNROLL);  // uniform scalar bound

  dot_partial_kernel<<<nblocks, THREADS, 0, stream>>>(
      (const float4*)A, (const float4*)B, partials, nvec, iters);

  const float invN = 1.0f / (float)n_rows;
  finalize_kernel<<<1, THREADS, 0, stream>>>(partials, nblocks, flip, out, invN);
}